// Linker_36344013259255
// MI455X (gfx1250) — compile-verified
//
#include <hip/hip_runtime.h>

// Problem constants (from reference): L=4, N=2048, DIM=512, B=64, NLINKS=512, TAU=1, iters=8
#define N_ROWS   2048
#define DIM_K    512
#define NB       64
#define NL       512
#define NUM_ITERS 8

typedef __attribute__((ext_vector_type(2))) float v2f;
typedef __attribute__((ext_vector_type(8))) float v8f;

// ---------------------------------------------------------------------------
// Kernel 1: gather + scaled batched GEMM via V_WMMA_F32_16X16X4_F32.
// Block = 256 threads (8 wave32), computes a 64x64 tile of scores[b].
// ---------------------------------------------------------------------------
__global__ __launch_bounds__(256) void scores_wmma_kernel(
    const float* __restrict__ reprs, const float* __restrict__ weight,
    const int* __restrict__ levels, const int* __restrict__ positions,
    float* __restrict__ out)
{
    __shared__ float At[64][66];   // pitch 66 floats -> bank stride 2, conflict-free
    __shared__ float Bt[64][66];
    __shared__ unsigned aOff[64];
    __shared__ unsigned bOff[64];

    const int b  = blockIdx.z;
    const int i0 = blockIdx.x * 64;
    const int j0 = blockIdx.y * 64;
    const int t  = threadIdx.x;

    // Resolve gathered row offsets (element units) once per block.
    if (t < 64) {
        const int i   = i0 + t;
        const int lvl = levels   [(b * NL + i) * 2 + 0];
        const int pos = positions[(b * NL + i) * 2 + 0];
        aOff[t] = (unsigned)(lvl * N_ROWS + pos) * DIM_K;
    } else if (t < 128) {
        const int rr  = t - 64;
        const int j   = j0 + rr;
        const int lvl = levels   [(b * NL + j) * 2 + 1];
        const int pos = positions[(b * NL + j) * 2 + 1];
        bOff[rr] = (unsigned)(lvl * N_ROWS + pos) * DIM_K;
    }
    __syncthreads();

    const int wave = t >> 5;
    const int lane = t & 31;
    const int half = lane >> 4;        // 0: K 0..1 / M 0..7 side, 1: K 2..3 / M 8..15 side
    const int l15  = lane & 15;
    const int wr   = wave & 3;         // 16-row sub-block 0..3
    const int wc0  = (wave >> 2) * 2;  // two 16-col sub-blocks per wave
    const int wc1  = wc0 + 1;

    v8f acc0 = {};
    v8f acc1 = {};

    const int row = t >> 2;            // staging: 4 threads per tile row
    const int seg = t & 3;

    for (int k0 = 0; k0 < DIM_K; k0 += 64) {
        // Stage A (pre-scaled by weight) and B tiles: 64 rows x 64 K.
        #pragma unroll
        for (int q = 0; q < 4; ++q) {
            const int col = seg * 16 + q * 4;
            const float4 wv = *(const float4*)&weight[k0 + col];
            const float4 av = *(const float4*)&reprs[aOff[row] + k0 + col];
            const float4 bv = *(const float4*)&reprs[bOff[row] + k0 + col];
            At[row][col + 0] = av.x * wv.x;
            At[row][col + 1] = av.y * wv.y;
            At[row][col + 2] = av.z * wv.z;
            At[row][col + 3] = av.w * wv.w;
            Bt[row][col + 0] = bv.x;
            Bt[row][col + 1] = bv.y;
            Bt[row][col + 2] = bv.z;
            Bt[row][col + 3] = bv.w;
        }
        __syncthreads();

        const int arow  = wr  * 16 + l15;  // A: lane 0-15 -> M, in-lane VGPRs -> K
        const int bcol0 = wc0 * 16 + l15;  // B: lane 0-15 -> N, in-VGPR rows -> K
        const int bcol1 = wc1 * 16 + l15;
        #pragma unroll
        for (int kk = 0; kk < 64; kk += 4) {
            const int kb = kk + half * 2;  // lanes 16-31 carry K+2..K+3
            v2f a  = *(const v2f*)&At[arow][kb];
            v2f b0 = *(const v2f*)&Bt[bcol0][kb];
            v2f b1 = *(const v2f*)&Bt[bcol1][kb];
            acc0 = __builtin_amdgcn_wmma_f32_16x16x4_f32(
                false, a, false, b0, (short)0, acc0, false, false);
            acc1 = __builtin_amdgcn_wmma_f32_16x16x4_f32(
                false, a, false, b1, (short)0, acc1, false, false);
        }
        __syncthreads();
    }

    // D layout: VGPR v -> M = v (lanes 0-15) / v+8 (lanes 16-31), N = lane&15.
    #pragma unroll
    for (int v = 0; v < 8; ++v) {
        const int    gi   = i0 + wr * 16 + v + half * 8;
        const size_t base = ((size_t)b * NL + gi) * NL;
        out[base + j0 + wc0 * 16 + l15] = acc0[v];  // TAU == 1.0
        out[base + j0 + wc1 * 16 + l15] = acc1[v];
    }
}

// ---------------------------------------------------------------------------
// Sinkhorn with lazy offsets: stored x never changes; r[b,i], c[b,j] evolve.
//   row step: r[i] = lse_j(x[i,j] - c[j])
//   col step: c[j] = lse_i(x[i,j] - r[i])
//   final:    out  = x - r[i] - c[j]
// ---------------------------------------------------------------------------

__global__ __launch_bounds__(256) void row_lse_kernel(
    const float* __restrict__ x, const float* __restrict__ c,
    float* __restrict__ r)
{
    const int row  = blockIdx.x * 8 + (threadIdx.x >> 5);  // global row = b*512+i
    const int lane = threadIdx.x & 31;
    const int b    = row >> 9;
    const float* xr = x + (size_t)row * NL;
    const float* cb = c + b * NL;

    float v[16];
    float m = -3.0e38f;
    #pragma unroll
    for (int q = 0; q < 16; ++q) {
        const int j = q * 32 + lane;
        v[q] = xr[j] - cb[j];
        m = fmaxf(m, v[q]);
    }
    #pragma unroll
    for (int off = 16; off >= 1; off >>= 1)
        m = fmaxf(m, __shfl_xor(m, off, 32));
    float s = 0.0f;
    #pragma unroll
    for (int q = 0; q < 16; ++q)
        s += __expf(v[q] - m);
    #pragma unroll
    for (int off = 16; off >= 1; off >>= 1)
        s += __shfl_xor(s, off, 32);
    if (lane == 0) r[row] = m + __logf(s);
}

__global__ __launch_bounds__(256) void col_lse_kernel(
    const float* __restrict__ x, const float* __restrict__ r,
    float* __restrict__ c)
{
    const int b = blockIdx.y;
    const int j = blockIdx.x * 256 + threadIdx.x;
    const float* xb = x + (size_t)b * NL * NL + j;
    const float* rb = r + b * NL;        // uniform across threads -> scalar loads

    float m = -3.0e38f, s = 0.0f;
    for (int i = 0; i < NL; ++i) {       // coalesced across threads each step
        const float v  = xb[(size_t)i * NL] - rb[i];
        const float m2 = fmaxf(m, v);
        s = s * __expf(m - m2) + __expf(v - m2);
        m = m2;
    }
    c[b * NL + j] = m + __logf(s);
}

__global__ __launch_bounds__(256) void apply_kernel(
    float* __restrict__ x, const float* __restrict__ r,
    const float* __restrict__ c)
{
    const int id  = blockIdx.x * 256 + threadIdx.x;  // one float4 each
    const int b   = id >> 16;
    const int rem = id & 65535;
    const int i   = rem >> 7;
    const int j4  = (rem & 127) * 4;
    const size_t base = (((size_t)b * NL) + i) * NL + j4;

    float4 xv = *(const float4*)(x + base);
    const float4 cv = *(const float4*)(c + b * NL + j4);
    const float  rv = r[b * NL + i];
    xv.x -= rv + cv.x;
    xv.y -= rv + cv.y;
    xv.z -= rv + cv.z;
    xv.w -= rv + cv.w;
    *(float4*)(x + base) = xv;
}

__global__ void zero_kernel(float* __restrict__ p, int n)
{
    const int id = blockIdx.x * blockDim.x + threadIdx.x;
    if (id < n) p[id] = 0.0f;
}

// ---------------------------------------------------------------------------
extern "C" void kernel_launch(void* const* d_in, const int* in_sizes, int n_in,
                              void* d_out, int out_size, void* d_ws, size_t ws_size,
                              hipStream_t stream)
{
    const float* reprs     = (const float*)d_in[0];
    const float* weight    = (const float*)d_in[1];
    const int*   levels    = (const int*)  d_in[2];
    const int*   positions = (const int*)  d_in[3];
    // d_in[4] = num_iters (== 8, compile-time constant here)

    float* out = (float*)d_out;
    float* r   = (float*)d_ws;          // NB*NL floats
    float* c   = r + NB * NL;           // NB*NL floats  (total 256 KB of ws)

    zero_kernel<<<(2 * NB * NL + 255) / 256, 256, 0, stream>>>(r, 2 * NB * NL);

    scores_wmma_kernel<<<dim3(NL / 64, NL / 64, NB), 256, 0, stream>>>(
        reprs, weight, levels, positions, out);

    for (int it = 0; it < NUM_ITERS; ++it) {
        row_lse_kernel<<<(NB * NL) / 8, 256, 0, stream>>>(out, c, r);
        col_lse_kernel<<<dim3(NL / 256, NB), 256, 0, stream>>>(out, r, c);
    }

    apply_kernel<<<(NB * NL * NL / 4) / 256, 256, 0, stream>>>(out, r, c);
}